// ConvEncoder_55018531062659
// MI455X (gfx1250) — compile-verified
//
#include <hip/hip_runtime.h>

// Rate-1/2 K=7 convolutional encoder, G0=1011011 G1=1111001.
// c0[t] = m[t]^m[t-2]^m[t-3]^m[t-5]^m[t-6]
// c1[t] = m[t]^m[t-1]^m[t-2]^m[t-3]^m[t-6]
// out[b, 2t] = c0, out[b, 2t+1] = c1  (float32)

#define K_BITS   4096
#define THREADS  256
#define BITS_PER_THREAD (K_BITS / THREADS)   // 16

__global__ __launch_bounds__(THREADS) void conv_encode_kernel(
    const float* __restrict__ in, float* __restrict__ out) {
  __shared__ __attribute__((aligned(16))) float smem[K_BITS];

  const int tid = threadIdx.x;
  const float* g_row = in + (size_t)blockIdx.x * K_BITS;

  // ---- Stage the whole 16KiB message row into LDS via CDNA5 async copies ----
  // 4 x B128 per thread, lanes contiguous -> fully coalesced.
#pragma unroll
  for (int k = 0; k < 4; ++k) {
    const int off = k * (THREADS * 4) + tid * 4;        // float index, 16B aligned
    const float* gaddr = g_row + off;
    unsigned lds_addr = (unsigned)(size_t)(&smem[off]); // LDS byte offset (low 32 bits)
    asm volatile("global_load_async_to_lds_b128 %0, %1, off"
                 :: "v"(lds_addr), "v"(gaddr)
                 : "memory");
  }
  asm volatile("s_wait_asynccnt 0" ::: "memory");
  __syncthreads();

  float* o_row = out + (size_t)blockIdx.x * (size_t)(2 * K_BITS);

  // ---- Encode: strided t so each wave's float2 stores are contiguous ----
#pragma unroll
  for (int k = 0; k < BITS_PER_THREAD; ++k) {
    const int t = k * THREADS + tid;

    unsigned w[7];
#pragma unroll
    for (int j = 0; j < 7; ++j) {
      const int idx = t - 6 + j;                 // zero initial encoder state
      const float f = (idx >= 0) ? smem[idx] : 0.0f;
      w[j] = (unsigned)f;                        // inputs are exactly 0.0f / 1.0f
    }

    const unsigned c0 = w[6] ^ w[4] ^ w[3] ^ w[1] ^ w[0];
    const unsigned c1 = w[6] ^ w[5] ^ w[4] ^ w[3] ^ w[0];

    float2 pair;
    pair.x = (float)c0;
    pair.y = (float)c1;
    *reinterpret_cast<float2*>(o_row + 2 * t) = pair;  // 8B/lane, 256B/wave, coalesced
  }
}

extern "C" void kernel_launch(void* const* d_in, const int* in_sizes, int n_in,
                              void* d_out, int out_size, void* d_ws, size_t ws_size,
                              hipStream_t stream) {
  (void)n_in; (void)d_ws; (void)ws_size; (void)out_size;
  const float* in = (const float*)d_in[0];   // [B, 4096] of 0.0/1.0
  float* out = (float*)d_out;                // [B, 8192]
  const int B = in_sizes[0] / K_BITS;        // 2048
  conv_encode_kernel<<<dim3(B), dim3(THREADS), 0, stream>>>(in, out);
}